// PrePrompt_75496935129282
// MI455X (gfx1250) — compile-verified
//
#include <hip/hip_runtime.h>
#include <math.h>

// Problem constants (match reference)
constexpr int   G    = 2048;
constexpr int   S    = 127;
constexpr int   D    = 256;
constexpr float TEMP = 10.0f;
constexpr float EPS  = 1e-8f;

typedef __attribute__((ext_vector_type(2))) float v2f;
typedef __attribute__((ext_vector_type(4))) float v4f;
typedef __attribute__((ext_vector_type(8))) float v8f;

// ---------------------------------------------------------------------------
// Kernel 1: zero the segment-sum accumulators (2*G*D floats) and the output.
// ---------------------------------------------------------------------------
__global__ void k_zero(float* __restrict__ p, int n, float* __restrict__ out) {
    int i = blockIdx.x * blockDim.x + threadIdx.x;
    if (i < n) p[i] = 0.0f;
    if (i == 0) out[0] = 0.0f;
}

// ---------------------------------------------------------------------------
// Kernel 2: fused segment sums. One 64-thread block per input row; each
// thread handles a float4 slice of both logits rows and scatter-adds with
// hardware f32 atomics into the L2-resident G x D accumulators.
// 410 MB streamed once -> HBM-bound phase (~18 us at 23.3 TB/s).
// ---------------------------------------------------------------------------
__global__ void k_segsum(const float* __restrict__ lo, const float* __restrict__ lp,
                         const int* __restrict__ oi, const int* __restrict__ pi,
                         float* __restrict__ sacc, float* __restrict__ pacc) {
    const int row = blockIdx.x;
    const int c   = threadIdx.x * 4;
    const int o   = oi[row];
    const int q   = pi[row];
    v4f a = *reinterpret_cast<const v4f*>(lo + (size_t)row * D + c);
    v4f b = *reinterpret_cast<const v4f*>(lp + (size_t)row * D + c);
    float* sd = sacc + (size_t)o * D + c;
    float* pd = pacc + (size_t)q * D + c;
#pragma unroll
    for (int i = 0; i < 4; ++i) {
        unsafeAtomicAdd(sd + i, a[i]);   // global_atomic_add_f32
        unsafeAtomicAdd(pd + i, b[i]);
    }
}

// ---------------------------------------------------------------------------
// Kernel 3: per-row L2 norms, max'ed with EPS. One wave32 per row.
// ---------------------------------------------------------------------------
__global__ void k_norm(const float* __restrict__ acc, float* __restrict__ nrm) {
    const int wave = threadIdx.x >> 5;
    const int lane = threadIdx.x & 31;
    const int row  = blockIdx.x * 8 + wave;
    const float* r = acc + (size_t)row * D + lane * 8;
    v4f x = *reinterpret_cast<const v4f*>(r);
    v4f y = *reinterpret_cast<const v4f*>(r + 4);
    float s = 0.0f;
#pragma unroll
    for (int i = 0; i < 4; ++i) s += x[i] * x[i] + y[i] * y[i];
#pragma unroll
    for (int off = 16; off > 0; off >>= 1) s += __shfl_xor(s, off, 32);
    if (lane == 0) nrm[row] = fmaxf(sqrtf(s), EPS);
}

// ---------------------------------------------------------------------------
// Kernel 4: DOT = self (GxD) x pos^T (DxG) via V_WMMA_F32_16X16X4_F32.
// Each wave computes a 32x32 C macro-tile (2x2 accumulator tiles, operand
// reuse: 4 WMMAs per 4 float2 loads). All operands are L2-resident (2 MB).
//
// Fragment layouts (ISA 7.12.2, 16x16x4 f32):
//   A 16x4 : lane l -> row (l&15), K-pair = 2*(l>>4)      -> float2 @ d
//   B 4x16 : lane l -> col (l&15), K-pair = 2*(l>>4); B[k][n] = pos[n][d]
//   C 16x16: VGPR v -> M = v + 8*(l>>4), N = l&15
// ---------------------------------------------------------------------------
__global__ void __launch_bounds__(128)
k_gemm(const float* __restrict__ A, const float* __restrict__ B,
       float* __restrict__ C) {
    const int w    = blockIdx.x * 4 + (threadIdx.x >> 5);
    const int tM   = (w >> 6) * 32;          // G/32 = 64 tiles per dim
    const int tN   = (w & 63) * 32;
    const int lane = threadIdx.x & 31;
    const int r    = lane & 15;
    const int koff = (lane >> 4) * 2;

    const float* pa0 = A + (size_t)(tM + r)      * D;
    const float* pa1 = A + (size_t)(tM + 16 + r) * D;
    const float* pb0 = B + (size_t)(tN + r)      * D;
    const float* pb1 = B + (size_t)(tN + 16 + r) * D;

    v8f c00 = {}, c01 = {}, c10 = {}, c11 = {};

    for (int k = 0; k < D; k += 4) {
        const int d0 = k + koff;
        v2f a0 = *reinterpret_cast<const v2f*>(pa0 + d0);
        v2f a1 = *reinterpret_cast<const v2f*>(pa1 + d0);
        v2f b0 = *reinterpret_cast<const v2f*>(pb0 + d0);
        v2f b1 = *reinterpret_cast<const v2f*>(pb1 + d0);
        c00 = __builtin_amdgcn_wmma_f32_16x16x4_f32(false, a0, false, b0,
                                                    (short)0, c00, false, false);
        c01 = __builtin_amdgcn_wmma_f32_16x16x4_f32(false, a0, false, b1,
                                                    (short)0, c01, false, false);
        c10 = __builtin_amdgcn_wmma_f32_16x16x4_f32(false, a1, false, b0,
                                                    (short)0, c10, false, false);
        c11 = __builtin_amdgcn_wmma_f32_16x16x4_f32(false, a1, false, b1,
                                                    (short)0, c11, false, false);
    }

    const int mh = 8 * (lane >> 4);
    const int n  = lane & 15;
#pragma unroll
    for (int v = 0; v < 8; ++v) {
        const int m = v + mh;
        C[(size_t)(tM + m)      * G + (tN + n)]      = c00[v];
        C[(size_t)(tM + m)      * G + (tN + 16 + n)] = c01[v];
        C[(size_t)(tM + 16 + m) * G + (tN + n)]      = c10[v];
        C[(size_t)(tM + 16 + m) * G + (tN + 16 + n)] = c11[v];
    }
}

// ---------------------------------------------------------------------------
// Kernel 5: gather 1 positive + 127 negatives per g from DOT, cosine-sim,
// softmax-style loss, wave reduction, atomic mean into d_out.
// ---------------------------------------------------------------------------
__global__ void k_loss(const float* __restrict__ DOT, const float* __restrict__ ns,
                       const float* __restrict__ np, const int* __restrict__ neg,
                       float* __restrict__ out) {
    const int wave = threadIdx.x >> 5;
    const int lane = threadIdx.x & 31;
    const int g    = blockIdx.x * 8 + wave;
    const float na = ns[g];
    float num = 0.0f, den = 0.0f;
    for (int j = lane; j < S + 1; j += 32) {
        const int col  = (j == 0) ? g : neg[g * S + j - 1];
        const float nb = np[col];
        const float sim = DOT[(size_t)g * G + col] / (na * nb);
        const float e   = expf(sim) * (1.0f / TEMP);
        if (j == 0) num = e; else den += e;
    }
#pragma unroll
    for (int off = 16; off > 0; off >>= 1) den += __shfl_xor(den, off, 32);
    if (lane == 0) {
        const float res = -logf(num / den);
        unsafeAtomicAdd(out, res * (1.0f / (float)G));
    }
}

// ---------------------------------------------------------------------------
extern "C" void kernel_launch(void* const* d_in, const int* in_sizes, int n_in,
                              void* d_out, int out_size, void* d_ws, size_t ws_size,
                              hipStream_t stream) {
    const float* lo  = (const float*)d_in[0];   // logits_origin (N,D)
    const float* lp  = (const float*)d_in[1];   // logits_pos    (N,D)
    const int*   oi  = (const int*)d_in[2];     // ori_idx (N)
    const int*   pi  = (const int*)d_in[3];     // pos_idx (N)
    const int*   neg = (const int*)d_in[4];     // neg_idx (G,S)
    const int    N   = in_sizes[2];

    float* ws   = (float*)d_ws;
    float* sacc = ws;                         // G*D
    float* pacc = ws + (size_t)G * D;         // G*D
    float* ns   = ws + (size_t)2 * G * D;     // G
    float* np   = ns + G;                     // G
    float* DOT  = np + G;                     // G*G
    float* out  = (float*)d_out;

    k_zero  <<<(2 * G * D + 255) / 256, 256, 0, stream>>>(sacc, 2 * G * D, out);
    k_segsum<<<N, 64, 0, stream>>>(lo, lp, oi, pi, sacc, pacc);
    k_norm  <<<G / 8, 256, 0, stream>>>(sacc, ns);
    k_norm  <<<G / 8, 256, 0, stream>>>(pacc, np);
    k_gemm  <<<(64 * 64) / 4, 128, 0, stream>>>(sacc, pacc, DOT);
    k_loss  <<<G / 8, 256, 0, stream>>>(DOT, ns, np, neg, out);
}